// Loss_17892833755500
// MI455X (gfx1250) — compile-verified
//
#include <hip/hip_runtime.h>

#define B_  64
#define NC_ 30
#define N_  8400
#define T_  32
#define PRED_STRIDE (34 * N_)      // floats per batch in pred
#define PC_ELEMS    (NC_ * N_)     // 252000 class logits per batch
#define PC_VEC4     (PC_ELEMS / 4) // 63000 float4 chunks per batch

typedef __attribute__((ext_vector_type(2))) float v2f;
typedef __attribute__((ext_vector_type(8))) float v8f;

__device__ __forceinline__ float softplus_f(float x) {
    // max(x,0) + log1p(exp(-|x|))  (numerically-stable log(1+e^x))
    float t = __expf(-fabsf(x));            // v_exp_f32 path
    return fmaxf(x, 0.0f) + __logf(1.0f + t); // v_log_f32 path
}

__global__ void init_out(float* out) {
    if (threadIdx.x == 0) out[0] = 0.0f;
}

__global__ __launch_bounds__(256) void bce_bulk(const float* __restrict__ pred,
                                                float* __restrict__ out) {
    const int b = blockIdx.y;
    const float4* __restrict__ pc4 =
        reinterpret_cast<const float4*>(pred + (size_t)b * PRED_STRIDE + 4 * N_);

    float acc = 0.0f;
    const int stride = gridDim.x * blockDim.x;
    for (int i = blockIdx.x * blockDim.x + threadIdx.x; i < PC_VEC4; i += stride) {
        float4 v = pc4[i];                   // global_load_b128, streamed once
        acc += softplus_f(v.x);
        acc += softplus_f(v.y);
        acc += softplus_f(v.z);
        acc += softplus_f(v.w);
    }
    acc *= (1.0f / (float)PC_ELEMS);         // mean over (NC, N)

    // --- wave32 reduction via V_WMMA_F32_16X16X4_F32 ---
    // A-vgpr0 = acc, A-vgpr1 = 0  =>  A[m,0]=acc[lane m], A[m,2]=acc[lane m+16]
    // B = all ones (4x16)         =>  D[m,n] = acc[m] + acc[m+16]  (all n equal)
    v2f a;  a[0]  = acc;  a[1]  = 0.0f;
    v2f bm; bm[0] = 1.0f; bm[1] = 1.0f;
    v8f c = {};
    c = __builtin_amdgcn_wmma_f32_16x16x4_f32(
            /*neg_a=*/false, a, /*neg_b=*/false, bm,
            /*c_mod=*/(short)0, c, /*reuse_a=*/false, /*reuse_b=*/false);
    // lanes 0..15 hold D[0..7, lane]  -> sum of lanes {0..7, 16..23}
    // lanes 16..31 hold D[8..15, l-16]-> sum of lanes {8..15, 24..31}
    float lanesum = c[0] + c[1] + c[2] + c[3] + c[4] + c[5] + c[6] + c[7];
    lanesum += __shfl_xor(lanesum, 16, 32);  // total over all 32 lanes

    __shared__ float wsum[8];
    const int lane = threadIdx.x & 31;
    const int wave = threadIdx.x >> 5;
    if (lane == 0) wsum[wave] = lanesum;
    __syncthreads();
    if (threadIdx.x == 0) {
        float s = 0.0f;
        #pragma unroll
        for (int w = 0; w < 8; ++w) s += wsum[w];
        atomicAdd(out, s);                   // one f32 atomic per block
    }
}

__global__ __launch_bounds__(256) void tgt_kernel(const float* __restrict__ pred,
                                                  const float* __restrict__ tboxes,
                                                  const int* __restrict__ tcls,
                                                  const int* __restrict__ tidx,
                                                  float* __restrict__ out) {
    const int g = blockIdx.x * blockDim.x + threadIdx.x;
    float contrib = 0.0f;
    if (g < B_ * T_) {
        const int b = g / T_;
        const int t = g - b * T_;
        const int my_idx = tidx[b * T_ + t];
        const int my_cls = tcls[b * T_ + t];

        // .at[].set(1.0) semantics: duplicate (cls,idx) within a batch counts once
        bool dup = false;
        for (int u = 0; u < t; ++u) {
            if (tidx[b * T_ + u] == my_idx && tcls[b * T_ + u] == my_cls) { dup = true; break; }
        }
        const float* pb = pred + (size_t)b * PRED_STRIDE;
        if (!dup)
            contrib -= pb[(4 + my_cls) * N_ + my_idx] * (1.0f / (float)PC_ELEMS);

        // MSE: gathered pred boxes vs targets*IMG_SIZE, mean over (T,4)=128
        float s = 0.0f;
        #pragma unroll
        for (int cdim = 0; cdim < 4; ++cdim) {
            float gv  = pb[cdim * N_ + my_idx];
            float box = tboxes[(b * T_ + t) * 4 + cdim] * 320.0f;
            float d = gv - box;
            s += d * d;
        }
        contrib += s * (1.0f / 128.0f);
    }

    // wave + block reduction
    #pragma unroll
    for (int off = 16; off >= 1; off >>= 1) contrib += __shfl_xor(contrib, off, 32);
    __shared__ float wsum[8];
    const int lane = threadIdx.x & 31;
    const int wave = threadIdx.x >> 5;
    if (lane == 0) wsum[wave] = contrib;
    __syncthreads();
    if (threadIdx.x == 0) {
        float s = 0.0f;
        #pragma unroll
        for (int w = 0; w < 8; ++w) s += wsum[w];
        atomicAdd(out, s);
    }
}

extern "C" void kernel_launch(void* const* d_in, const int* in_sizes, int n_in,
                              void* d_out, int out_size, void* d_ws, size_t ws_size,
                              hipStream_t stream) {
    const float* pred   = (const float*)d_in[0];
    const float* tboxes = (const float*)d_in[1];
    const int*   cls    = (const int*)d_in[2];
    const int*   idx    = (const int*)d_in[3];
    float* out = (float*)d_out;

    init_out<<<1, 32, 0, stream>>>(out);

    dim3 grid(64, B_);                       // 4096 blocks; ~4 float4s/thread/batch
    bce_bulk<<<grid, 256, 0, stream>>>(pred, out);

    tgt_kernel<<<(B_ * T_ + 255) / 256, 256, 0, stream>>>(pred, tboxes, cls, idx, out);
}